// PhaseGradientRedistributor_34110630265364
// MI455X (gfx1250) — compile-verified
//
#include <hip/hip_runtime.h>
#include <hip/hip_bf16.h>

typedef float v2f __attribute__((ext_vector_type(2)));
typedef float v8f __attribute__((ext_vector_type(8)));

#define N 8192
#define ROWS_PER_BLOCK 16
#define WAVES 8
#define CHUNK 64              // columns per tile
#define NCHUNK (N / CHUNK)    // 128
#define LDS_PITCH 68          // 16x64 tile padded; 68*4B = 272B, 16B aligned, conflict-free

// ---------------- kernel 1: zero the scalar accumulators ----------------
__global__ void k_init(float* acc) {
    if (threadIdx.x < 4) acc[threadIdx.x] = 0.0f;
}

// ---------------- kernel 2: per-j vectors + mean sin/cos ----------------
// acc[0] += sum cos(p), acc[1] += sum sin(p); su[j]=a_j sin p_j, cu[j]=a_j cos p_j
__global__ __launch_bounds__(256) void k_prepass(const float* __restrict__ phases,
                                                 const float* __restrict__ alive,
                                                 float* __restrict__ su,
                                                 float* __restrict__ cu,
                                                 float* acc) {
    int i = blockIdx.x * 256 + threadIdx.x;
    float p = phases[i];
    float s = sinf(p), c = cosf(p);
    float a = alive[i];
    su[i] = a * s;
    cu[i] = a * c;
    // block reduction of raw sin/cos for R
    float ss = s, cs = c;
    for (int off = 16; off > 0; off >>= 1) {
        ss += __shfl_down(ss, off, 32);
        cs += __shfl_down(cs, off, 32);
    }
    __shared__ float sl[WAVES], cl[WAVES];
    int wave = threadIdx.x >> 5, lane = threadIdx.x & 31;
    if (lane == 0) { sl[wave] = ss; cl[wave] = cs; }
    __syncthreads();
    if (threadIdx.x == 0) {
        float S = 0.f, C = 0.f;
        for (int w = 0; w < WAVES; ++w) { S += sl[w]; C += cl[w]; }
        atomicAdd(&acc[0], C);
        atomicAdd(&acc[1], S);
    }
}

// ---------------- kernel 3: main bandwidth-bound pass ----------------
// Each block: 16 rows. Each wave streams 16x64 tiles of w = K*dist (coalesced
// float4 loads), fuses sum(K^2)/sum(|K|), and reduces rows via
// V_WMMA_F32_16X16X4_F32 with A = [su; cu; 0...] (2x4 slab), B = w-tile (4x16).
__global__ __launch_bounds__(256) void k_coupling(const float* __restrict__ phases,
                                                  const float* __restrict__ alive,
                                                  const float* __restrict__ dist,
                                                  const float* __restrict__ K,
                                                  const float* __restrict__ su,
                                                  const float* __restrict__ cu,
                                                  float* __restrict__ out,
                                                  float* acc) {
    const int i0   = blockIdx.x * ROWS_PER_BLOCK;
    const int tid  = threadIdx.x;
    const int wave = tid >> 5;
    const int lane = tid & 31;
    const int lhi  = lane >> 4;   // 0 for lanes 0-15, 1 for lanes 16-31
    const int llo  = lane & 15;

    __shared__ float wlds[WAVES][ROWS_PER_BLOCK][LDS_PITCH]; // ~34.8 KB
    __shared__ float sulds[WAVES][CHUNK];
    __shared__ float culds[WAVES][CHUNK];
    __shared__ float sred[WAVES][ROWS_PER_BLOCK];
    __shared__ float cred[WAVES][ROWS_PER_BLOCK];
    __shared__ float k2s[WAVES], kabss[WAVES];

    float* wl = &wlds[wave][0][0];
    v8f cacc = {};                 // D[0,n]=S_n partial, D[1,n]=C_n partial
    float k2 = 0.0f, kabs = 0.0f;

    for (int ch = wave; ch < NCHUNK; ch += WAVES) {
        const int j0 = ch * CHUNK;

        // stage sin/cos slab for this column chunk (coalesced, tiny)
        sulds[wave][lane]      = su[j0 + lane];
        sulds[wave][lane + 32] = su[j0 + lane + 32];
        culds[wave][lane]      = cu[j0 + lane];
        culds[wave][lane + 32] = cu[j0 + lane + 32];

        // stream 16x64 tile: each iter a half-wave reads 256B contiguous
        const bool pf = (ch + WAVES) < NCHUNK;
#pragma unroll
        for (int rr = 0; rr < 8; ++rr) {
            const int row = rr * 2 + lhi;
            const int col = llo * 4;
            const size_t g = (size_t)(i0 + row) * N + (size_t)(j0 + col);
            if (pf) {   // prefetch this wave's next tile (j0 + 512)
                __builtin_prefetch(K + g + (size_t)(WAVES * CHUNK), 0, 0);
                __builtin_prefetch(dist + g + (size_t)(WAVES * CHUNK), 0, 0);
            }
            const float4 kv = *reinterpret_cast<const float4*>(K + g);
            const float4 dv = *reinterpret_cast<const float4*>(dist + g);
            k2   += kv.x * kv.x + kv.y * kv.y + kv.z * kv.z + kv.w * kv.w;
            kabs += fabsf(kv.x) + fabsf(kv.y) + fabsf(kv.z) + fabsf(kv.w);
            float4 w;
            w.x = kv.x * dv.x; w.y = kv.y * dv.y;
            w.z = kv.z * dv.z; w.w = kv.w * dv.w;
            *reinterpret_cast<float4*>(&wl[row * LDS_PITCH + col]) = w;
        }
        // wave-private LDS tile: DS ops are in-order per wave; fence compiler + counters
        __builtin_amdgcn_wave_barrier();
        asm volatile("s_wait_dscnt 0" ::: "memory");

        // 16 WMMA steps cover K=64 columns
#pragma unroll
        for (int kk = 0; kk < 16; ++kk) {
            const int kb = kk * 4 + lhi * 2;   // this lane-half's K pair
            // A fragment (16x4): lane llo = row M; only M=0 (su) and M=1 (cu) nonzero
            const v2f sv = *reinterpret_cast<const v2f*>(&sulds[wave][kb]);
            const v2f cv = *reinterpret_cast<const v2f*>(&culds[wave][kb]);
            v2f a;
            a.x = (llo == 0) ? sv.x : ((llo == 1) ? cv.x : 0.0f);
            a.y = (llo == 0) ? sv.y : ((llo == 1) ? cv.y : 0.0f);
            // B fragment (4x16): lane llo = column N = row index; K pair contiguous in LDS row
            const v2f b = *reinterpret_cast<const v2f*>(&wl[llo * LDS_PITCH + kb]);
            cacc = __builtin_amdgcn_wmma_f32_16x16x4_f32(
                false, a, false, b, (short)0, cacc, false, false);
        }
        __builtin_amdgcn_wave_barrier();
    }

    // per-wave partials: lane n (0-15): VGPR0 = D[0,n] = S_n, VGPR1 = D[1,n] = C_n
    if (lhi == 0) {
        sred[wave][llo] = cacc[0];
        cred[wave][llo] = cacc[1];
    }

    // fused sum(K^2), sum(|K|): wave shuffle reduce, then block combine
    for (int off = 16; off > 0; off >>= 1) {
        k2   += __shfl_down(k2, off, 32);
        kabs += __shfl_down(kabs, off, 32);
    }
    if (lane == 0) { k2s[wave] = k2; kabss[wave] = kabs; }
    __syncthreads();

    if (tid == 0) {
        float a2 = 0.f, aa = 0.f;
        for (int w = 0; w < WAVES; ++w) { a2 += k2s[w]; aa += kabss[w]; }
        atomicAdd(&acc[2], a2);
        atomicAdd(&acc[3], aa);
    }
    if (tid < ROWS_PER_BLOCK) {
        float S = 0.f, C = 0.f;
        for (int w = 0; w < WAVES; ++w) { S += sred[w][tid]; C += cred[w][tid]; }
        const int i = i0 + tid;
        const float p = phases[i];
        out[1 + i] = alive[i] * (cosf(p) * S - sinf(p) * C);
    }
}

// ---------------- kernel 4: R and loss ----------------
__global__ void k_finalize(const float* acc, float* out) {
    if (threadIdx.x == 0 && blockIdx.x == 0) {
        const float mc = acc[0] / (float)N;
        const float ms = acc[1] / (float)N;
        const float R = sqrtf(mc * mc + ms * ms);
        out[0] = R;
        out[1 + N] = 1.0f - R + 0.01f * sqrtf(acc[2]) + 0.01f * acc[3];
    }
}

extern "C" void kernel_launch(void* const* d_in, const int* in_sizes, int n_in,
                              void* d_out, int out_size, void* d_ws, size_t ws_size,
                              hipStream_t stream) {
    const float* phases = (const float*)d_in[0];   // [N]
    const float* alive  = (const float*)d_in[1];   // [N]
    const float* dist   = (const float*)d_in[2];   // [N,N]
    const float* K      = (const float*)d_in[3];   // [N,N]
    float* out = (float*)d_out;                    // [0]=R, [1..N]=dtheta, [N+1]=loss

    float* ws  = (float*)d_ws;
    float* acc = ws;            // 4 scalar accumulators (16-float pad)
    float* su  = ws + 16;       // [N]
    float* cu  = ws + 16 + N;   // [N]

    k_init<<<1, 32, 0, stream>>>(acc);
    k_prepass<<<N / 256, 256, 0, stream>>>(phases, alive, su, cu, acc);
    k_coupling<<<N / ROWS_PER_BLOCK, 256, 0, stream>>>(phases, alive, dist, K, su, cu, out, acc);
    k_finalize<<<1, 32, 0, stream>>>(acc, out);
}